// GraphConvLayer_48232482734926
// MI455X (gfx1250) — compile-verified
//
#include <hip/hip_runtime.h>

typedef __attribute__((ext_vector_type(2))) float v2f;
typedef __attribute__((ext_vector_type(8))) float v8f;

#define DIMF 128
#define LDS_STRIDE 132   // 128 + 4 pad: keeps float4 alignment, kills bank conflicts

// ---------------- init / degree kernels ----------------

__global__ void gcn_zero_f4(float4* __restrict__ p, long long n4) {
    long long i = (long long)blockIdx.x * blockDim.x + threadIdx.x;
    if (i < n4) p[i] = make_float4(0.f, 0.f, 0.f, 0.f);
}

__global__ void gcn_deg_init(float* __restrict__ deg, int n) {
    int i = blockIdx.x * blockDim.x + threadIdx.x;
    if (i < n) deg[i] = 1.0f;   // self-loop contribution
}

__global__ void gcn_deg_count(const int* __restrict__ cols, float* __restrict__ deg, int e) {
    int i = blockIdx.x * blockDim.x + threadIdx.x;
    if (i < e) atomicAdd(&deg[cols[i]], 1.0f);
}

__global__ void gcn_deg_to_dinv(float* __restrict__ deg, int n) {
    int i = blockIdx.x * blockDim.x + threadIdx.x;
    if (i < n) deg[i] = rsqrtf(deg[i]);   // deg >= 1 always (self-loops)
}

// ---------------- WMMA GEMM: xw = x @ W ----------------
// Block = 256 threads = 8 waves. Block computes rows [16*blk, 16*blk+16) x all 128 cols.
// Wave w owns output cols [16w, 16w+16). K=128 consumed in 32 steps of
// v_wmma_f32_16x16x4_f32.

__global__ __launch_bounds__(256) void gcn_gemm_xw(const float* __restrict__ x,
                                                   const float* __restrict__ W,
                                                   float* __restrict__ xw, int n) {
    __shared__ float ax[16 * LDS_STRIDE];

    const int t  = threadIdx.x;
    const int m0 = blockIdx.x * 16;

    // Stage 16x128 x-tile into LDS, coalesced float4 loads (512 float4s / 256 threads).
    #pragma unroll
    for (int i = t; i < 512; i += 256) {
        int row = i >> 5;        // 0..15
        int kq  = i & 31;        // float4 column
        int gr  = m0 + row;
        float4 v = make_float4(0.f, 0.f, 0.f, 0.f);
        if (gr < n) v = ((const float4*)(x + (long long)gr * DIMF))[kq];
        *(float4*)&ax[row * LDS_STRIDE + kq * 4] = v;
    }
    __syncthreads();

    const int lane    = t & 31;
    const int wave    = t >> 5;
    const int colbase = wave * 16;
    const int ln16    = lane & 15;        // M for A, N for B/D
    const int kb      = (lane >> 4) * 2;  // half-wave K pair select

    v8f acc = {};
    #pragma unroll
    for (int k0 = 0; k0 < DIMF; k0 += 4) {
        v2f a, b;
        // A fragment (16x4 f32): lanes 0-15 hold K={k0,k0+1}, lanes 16-31 K={k0+2,k0+3}
        a.x = ax[ln16 * LDS_STRIDE + k0 + kb];
        a.y = ax[ln16 * LDS_STRIDE + k0 + kb + 1];
        // B fragment (4x16 f32): row K striped across lanes (N = lane&15)
        b.x = W[(k0 + kb) * DIMF + colbase + ln16];
        b.y = W[(k0 + kb + 1) * DIMF + colbase + ln16];
        acc = __builtin_amdgcn_wmma_f32_16x16x4_f32(
            /*neg_a=*/false, a, /*neg_b=*/false, b,
            /*c_mod=*/(short)0, acc, /*reuse_a=*/false, /*reuse_b=*/false);
    }

    // D layout: VGPR v -> M = v + 8*(lane>>4), N = lane&15
    #pragma unroll
    for (int v = 0; v < 8; v++) {
        int gr = m0 + v + (lane >> 4) * 8;
        if (gr < n) xw[(long long)gr * DIMF + colbase + ln16] = acc[v];
    }
}

// ---------------- edge scatter: out[col] += xw[row] * dinv[row]*dinv[col] ----
// One wave per edge; 32 lanes x float4 = 128 features.

__global__ __launch_bounds__(256) void gcn_edge_scatter(const int* __restrict__ rows,
                                                        const int* __restrict__ cols,
                                                        const float* __restrict__ dinv,
                                                        const float* __restrict__ xw,
                                                        float* __restrict__ out, int e) {
    int eid  = blockIdx.x * 8 + (threadIdx.x >> 5);
    int lane = threadIdx.x & 31;
    if (eid >= e) return;
    int r = rows[eid];
    int c = cols[eid];
    float nrm = dinv[r] * dinv[c];
    float4 v = ((const float4*)(xw + (long long)r * DIMF))[lane];
    float* dst = out + (long long)c * DIMF + lane * 4;
    atomicAdd(dst + 0, v.x * nrm);
    atomicAdd(dst + 1, v.y * nrm);
    atomicAdd(dst + 2, v.z * nrm);
    atomicAdd(dst + 3, v.w * nrm);
}

// ---------------- epilogue: out = relu(acc + xw*dinv^2 + b) ----------------

__global__ void gcn_epilogue(float* __restrict__ out, const float* __restrict__ xw,
                             const float* __restrict__ dinv, const float* __restrict__ b,
                             int n) {
    long long i = (long long)blockIdx.x * blockDim.x + threadIdx.x;
    long long total = (long long)n * 32;   // float4 chunks
    if (i >= total) return;
    int node = (int)(i >> 5);
    int q    = (int)(i & 31);
    float d  = dinv[node];
    float s  = d * d;   // self-loop norm
    float4 acc = ((const float4*)(out + (long long)node * DIMF))[q];
    float4 xv  = ((const float4*)(xw  + (long long)node * DIMF))[q];
    float4 bv  = ((const float4*)b)[q];
    float4 r;
    r.x = fmaxf(acc.x + xv.x * s + bv.x, 0.f);
    r.y = fmaxf(acc.y + xv.y * s + bv.y, 0.f);
    r.z = fmaxf(acc.z + xv.z * s + bv.z, 0.f);
    r.w = fmaxf(acc.w + xv.w * s + bv.w, 0.f);
    ((float4*)(out + (long long)node * DIMF))[q] = r;
}

// ---------------- launcher ----------------

extern "C" void kernel_launch(void* const* d_in, const int* in_sizes, int n_in,
                              void* d_out, int out_size, void* d_ws, size_t ws_size,
                              hipStream_t stream) {
    const float* x    = (const float*)d_in[0];
    const int*   edge = (const int*)d_in[1];
    const float* W    = (const float*)d_in[2];
    const float* bias = (const float*)d_in[3];

    const int n = in_sizes[0] / DIMF;
    const int e = in_sizes[1] / 2;
    const int* rows = edge;        // edge_index[0] = source
    const int* cols = edge + e;    // edge_index[1] = target

    float* out = (float*)d_out;

    // workspace: [deg/dinv : n floats (padded)] [xw : n*128 floats]
    float* deg = (float*)d_ws;
    long long npad = ((long long)n + 255) & ~255LL;
    float* xw = deg + npad;

    const long long n4 = (long long)n * 32;   // output float4 count

    gcn_zero_f4<<<(unsigned)((n4 + 255) / 256), 256, 0, stream>>>((float4*)out, n4);
    gcn_deg_init<<<(n + 255) / 256, 256, 0, stream>>>(deg, n);
    gcn_deg_count<<<(e + 255) / 256, 256, 0, stream>>>(cols, deg, e);
    gcn_deg_to_dinv<<<(n + 255) / 256, 256, 0, stream>>>(deg, n);
    gcn_gemm_xw<<<(n + 15) / 16, 256, 0, stream>>>(x, W, xw, n);
    gcn_edge_scatter<<<(e + 7) / 8, 256, 0, stream>>>(rows, cols, deg, xw, out, e);
    gcn_epilogue<<<(unsigned)((n4 + 255) / 256), 256, 0, stream>>>(out, xw, deg, bias, n);
}